// SparseMatmulGateUp_62878321214307
// MI455X (gfx1250) — compile-verified
//
#include <hip/hip_runtime.h>
#include <hip/hip_bf16.h>

// Problem constants (match reference).
#define BDIM 2048
#define EDIM 8
#define HDIM 2048
#define NDIM 8192

#define NCHUNK 8              // stage-1 split of N
#define HCHUNK 4              // stage-2 split of H

typedef float v2f __attribute__((ext_vector_type(2)));
typedef float v8f __attribute__((ext_vector_type(8)));

// ---------------------------------------------------------------------------
// Init: zero wsum (E*H floats in workspace) and out (B floats). Both buffers
// are poisoned by the harness, so we must clear them every call.
// ---------------------------------------------------------------------------
__global__ void zero_init_kernel(float* __restrict__ wsum, float* __restrict__ out) {
    int i = blockIdx.x * blockDim.x + threadIdx.x;
    if (i < EDIM * HDIM) wsum[i] = 0.0f;
    if (i < BDIM)        out[i]  = 0.0f;
}

// ---------------------------------------------------------------------------
// Stage 1: wsum[e,h] = sum_n weight[e,h,n].
// One wave (32 threads) per (16-row group, n-chunk).
// WMMA f32 16x16x4 with B = all-ones performs a 16-row x 4-col reduction per
// instruction; D accumulates row sums replicated across all 16 columns.
// A-matrix f32 16x4 layout: lane L holds row (L%16); VGPR0/1 = K{0,1} for
// lanes 0-15, K{2,3} for lanes 16-31  ->  one b64 load per lane per wmma.
// ---------------------------------------------------------------------------
__global__ void rowsum_wmma_kernel(const float* __restrict__ W,
                                   float* __restrict__ wsum) {
    const int lane  = threadIdx.x;        // 0..31
    const int half  = lane >> 4;          // 0 or 1
    const int m     = lane & 15;          // row within group
    const int row0  = blockIdx.x * 16;    // row group base (rows = e*H + h)
    const int nbase = blockIdx.y * (NDIM / NCHUNK);

    const float* p = W + (long long)(row0 + m) * NDIM + nbase + half * 2;

    v8f c0 = {0.f, 0.f, 0.f, 0.f, 0.f, 0.f, 0.f, 0.f};
    v8f c1 = c0;
    const v2f ones = {1.0f, 1.0f};

    const int steps = (NDIM / NCHUNK) / 4;   // 256 wmma steps, even
#pragma unroll 4
    for (int s = 0; s < steps; s += 2) {
        v2f a0 = *(const v2f*)(p);
        v2f a1 = *(const v2f*)(p + 4);
        p += 8;
        c0 = __builtin_amdgcn_wmma_f32_16x16x4_f32(false, a0, false, ones,
                                                   (short)0, c0, false, false);
        c1 = __builtin_amdgcn_wmma_f32_16x16x4_f32(false, a1, false, ones,
                                                   (short)0, c1, false, false);
    }
    c0 += c1;

    // D layout: VGPR r, lanes 0-15 -> (M=r, N=lane); lanes 16-31 -> (M=r+8).
    // Lane 0 holds rows 0..7 in c0[0..7]; lane 16 holds rows 8..15.
    if (m == 0) {
        const int mbase = half * 8;
#pragma unroll
        for (int r = 0; r < 8; ++r)
            atomicAdd(&wsum[row0 + mbase + r], c0[r]);
    }
}

// ---------------------------------------------------------------------------
// Stage 2: out[b] += mask[b,e] * sum_h hidden[b,e,h] * wsum[e,h].
// One wave per (16-token group, expert e, h-chunk). A = hidden tile,
// B = wsum broadcast across all 16 columns (B f32 4x16 layout: column = lane%16,
// VGPR0/1 = K{0,1} for lanes 0-15, K{2,3} for lanes 16-31 -> per-lane b64).
// Mask applied once per (token, e) at the end, before the f32 atomic add.
// ---------------------------------------------------------------------------
__global__ void gemv_wmma_kernel(const float* __restrict__ Hid,
                                 const int* __restrict__ Msk,
                                 const float* __restrict__ wsum,
                                 float* __restrict__ out) {
    const int lane = threadIdx.x;
    const int half = lane >> 4;
    const int m    = lane & 15;
    const int t0   = blockIdx.x * 16;                 // token group base
    const int e    = blockIdx.y;                      // expert
    const int hb   = blockIdx.z * (HDIM / HCHUNK);    // h-chunk base

    const float* pa = Hid + ((long long)(t0 + m) * EDIM + e) * HDIM + hb + half * 2;
    const float* pb = wsum + e * HDIM + hb + half * 2;

    v8f c0 = {0.f, 0.f, 0.f, 0.f, 0.f, 0.f, 0.f, 0.f};
    v8f c1 = c0;

    const int steps = (HDIM / HCHUNK) / 4;   // 128 wmma steps, even
#pragma unroll 4
    for (int s = 0; s < steps; s += 2) {
        v2f a0 = *(const v2f*)(pa);
        v2f a1 = *(const v2f*)(pa + 4);
        v2f b0 = *(const v2f*)(pb);
        v2f b1 = *(const v2f*)(pb + 4);
        pa += 8;
        pb += 8;
        c0 = __builtin_amdgcn_wmma_f32_16x16x4_f32(false, a0, false, b0,
                                                   (short)0, c0, false, false);
        c1 = __builtin_amdgcn_wmma_f32_16x16x4_f32(false, a1, false, b1,
                                                   (short)0, c1, false, false);
    }
    c0 += c1;

    // Per-token dot products replicated across D columns; lanes 0 and 16
    // together hold all 16 token partials in c0[0..7].
    if (m == 0) {
        const int mbase = half * 8;
#pragma unroll
        for (int r = 0; r < 8; ++r) {
            const int t = t0 + mbase + r;
            if (Msk[t * EDIM + e] != 0)
                atomicAdd(&out[t], c0[r]);
        }
    }
}

// ---------------------------------------------------------------------------
// Launcher.
// d_in[0] = hidden_4d [B,E,1,H] f32, d_in[1] = sparsity [B,E] i32,
// d_in[2] = weight [1,E,H,N] f32.  d_out = [B] f32.  d_ws >= E*H*4 bytes.
// ---------------------------------------------------------------------------
extern "C" void kernel_launch(void* const* d_in, const int* in_sizes, int n_in,
                              void* d_out, int out_size, void* d_ws, size_t ws_size,
                              hipStream_t stream) {
    const float* hidden = (const float*)d_in[0];
    const int*   spars  = (const int*)d_in[1];
    const float* weight = (const float*)d_in[2];
    float* out  = (float*)d_out;
    float* wsum = (float*)d_ws;      // E*H floats = 64 KB

    // 1) zero accumulators (wsum + out)
    zero_init_kernel<<<(EDIM * HDIM + 255) / 256, 256, 0, stream>>>(wsum, out);

    // 2) wsum[e,h] = sum_n weight[e,h,n]   (streams the 537 MB weight once)
    rowsum_wmma_kernel<<<dim3((EDIM * HDIM) / 16, NCHUNK), 32, 0, stream>>>(weight, wsum);

    // 3) out[b] = sum_e mask * <hidden[b,e,:], wsum[e,:]>  (streams 134 MB hidden)
    gemv_wmma_kernel<<<dim3(BDIM / 16, EDIM, HCHUNK), 32, 0, stream>>>(hidden, spars, wsum, out);
}